// MultiHeadAttention_80418967650946
// MI455X (gfx1250) — compile-verified
//
#include <hip/hip_runtime.h>
#include <math.h>

#define B_   4
#define H_   16
#define D_   4
#define HID_ 256
#define N_   1024
#define SCALE_   0.5f      // D^-0.5, D=4
#define BN_EPS_  1e-5f
#define NEG_SLOPE_ 0.2f

typedef __attribute__((ext_vector_type(2))) float v2f;
typedef __attribute__((ext_vector_type(8))) float v8f;

__device__ __forceinline__ float hmax16(float v) {
    v = fmaxf(v, __shfl_xor(v, 1, 32));
    v = fmaxf(v, __shfl_xor(v, 2, 32));
    v = fmaxf(v, __shfl_xor(v, 4, 32));
    v = fmaxf(v, __shfl_xor(v, 8, 32));
    return v;
}
__device__ __forceinline__ float hsum16(float v) {
    v += __shfl_xor(v, 1, 32);
    v += __shfl_xor(v, 2, 32);
    v += __shfl_xor(v, 4, 32);
    v += __shfl_xor(v, 8, 32);
    return v;
}

// ---------------------------------------------------------------------------
// Kernel 1: 1x1-conv projections q/k/v -> permuted [B,H,N,4] layouts that
// resolve the reference's reshape(B,D,N,H) view:
//   h = n & 15 ; m = (o & 15)*64 + (n >> 4) ; d = o >> 4
// ---------------------------------------------------------------------------
__global__ void __launch_bounds__(256)
proj_kernel(const float* __restrict__ q, const float* __restrict__ k,
            const float* __restrict__ v,
            const float* __restrict__ Wq, const float* __restrict__ Wk,
            const float* __restrict__ Wv,
            float* __restrict__ Qh, float* __restrict__ KhT,
            float* __restrict__ Vh) {
    int t      = blockIdx.x;          // 0..3071
    int tensor = t >> 10;             // 0:q 1:k 2:v
    int r      = t & 1023;
    int b      = r >> 8;
    int o      = (r >> 2) & 63;
    int n      = ((r & 3) << 8) + threadIdx.x;

    const float* in = (tensor == 0) ? q  : (tensor == 1) ? k  : v;
    const float* W  = (tensor == 0) ? Wq : (tensor == 1) ? Wk : Wv;
    float*       ds = (tensor == 0) ? Qh : (tensor == 1) ? KhT : Vh;

    const float* ip = in + ((size_t)b * HID_) * N_ + n;   // q[b][c][n], c strided
    const float* wp = W + o * HID_;                       // uniform -> SMEM loads
    float acc = 0.0f;
#pragma unroll 8
    for (int c = 0; c < HID_; ++c)
        acc = fmaf(wp[c], ip[(size_t)c * N_], acc);

    int h = n & 15;
    int m = ((o & 15) << 6) | (n >> 4);
    int d = o >> 4;
    ds[(((size_t)b * H_ + h) * N_ + m) * D_ + d] = acc;
}

// ---------------------------------------------------------------------------
// Kernel 2: fused attention. One wave32 per 16-row tile of one (b,h).
// S(16x16) = (Q*scale)(16x4) x K^T(4x16) + bias  via V_WMMA_F32_16X16X4_F32
// (bias tile enters as the C operand). PER-LANE online softmax: each lane
// tracks (m,l,acc) for its own 64 columns -> ZERO cross-lane ops in the hot
// loop. Partial states merged once at the end (split-softmax merge).
// Bias (256 MB) streamed exactly once, non-temporal.
// ---------------------------------------------------------------------------
__global__ void __launch_bounds__(256)
attn_kernel(const float* __restrict__ Qh, const float* __restrict__ KhT,
            const float* __restrict__ Vh, const float* __restrict__ bias,
            float* __restrict__ X) {
    const int wave = threadIdx.x >> 5;
    const int lane = threadIdx.x & 31;
    const int hg   = lane >> 4;        // half-group: 0 -> rows r, 1 -> rows r+8
    const int lr   = lane & 15;

    const int tile    = blockIdx.x * 8 + wave;   // 0..4095
    const int bh      = tile >> 6;               // b*H + h
    const int rowTile = tile & 63;
    const int b = bh >> 4, h = bh & 15;

    // A fragment: 16x4 A-matrix layout, lanes 0-15 K={0,1}, lanes 16-31 K={2,3}
    const int qrow = rowTile * 16 + lr;
    const float4 qv = *(const float4*)(Qh + ((size_t)bh * N_ + qrow) * D_);
    v2f a;
    a[0] = (hg ? qv.z : qv.x) * SCALE_;
    a[1] = (hg ? qv.w : qv.y) * SCALE_;

    // bias pointer for this lane's C-layout position (row base rowTile*16+hg*8)
    const float* bias_row =
        bias + ((size_t)bh * N_ + rowTile * 16 + hg * 8) * N_ + lr;

    // per-lane softmax state over this lane's 64 columns, per row
    float m_ln[8], l_ln[8], acc[8][4];
#pragma unroll
    for (int r = 0; r < 8; ++r) {
        m_ln[r] = -1e30f;
        l_ln[r] = 0.0f;
        acc[r][0] = acc[r][1] = acc[r][2] = acc[r][3] = 0.0f;
    }

    for (int blk = 0; blk < 64; ++blk) {
        const int col = blk * 16 + lr;
        // B fragment: 4x16 B-matrix layout, columns across lanes
        const float4 kv = *(const float4*)(KhT + ((size_t)bh * N_ + col) * D_);
        v2f bf;
        bf[0] = hg ? kv.z : kv.x;
        bf[1] = hg ? kv.w : kv.y;

        // C operand = bias tile (non-temporal: 256MB streamed exactly once)
        v8f c;
#pragma unroll
        for (int r = 0; r < 8; ++r)
            c[r] = __builtin_nontemporal_load(bias_row + (size_t)blk * 16 +
                                              (size_t)r * N_);
        if (blk < 63)
            __builtin_prefetch(bias_row + (size_t)(blk + 1) * 16, 0, 0);

        // S = A x B + C   (EXEC all ones here; uniform control flow)
        c = __builtin_amdgcn_wmma_f32_16x16x4_f32(false, a, false, bf,
                                                  (short)0, c, false, false);

        const float4 vv = *(const float4*)(Vh + ((size_t)bh * N_ + col) * D_);
#pragma unroll
        for (int r = 0; r < 8; ++r) {
            const float s    = c[r];
            const float mnew = fmaxf(m_ln[r], s);
            const float sc   = __expf(m_ln[r] - mnew);   // 1.0 when max unchanged
            const float p    = __expf(s - mnew);
            m_ln[r] = mnew;
            l_ln[r] = fmaf(l_ln[r], sc, p);
            acc[r][0] = fmaf(p, vv.x, acc[r][0] * sc);
            acc[r][1] = fmaf(p, vv.y, acc[r][1] * sc);
            acc[r][2] = fmaf(p, vv.z, acc[r][2] * sc);
            acc[r][3] = fmaf(p, vv.w, acc[r][3] * sc);
        }
    }

    // merge the 16 per-lane partial softmax states per row, normalize, write
    // X[b][h*4+d][n]  (the transpose(3,2).view(B,H*D,N) layout)
#pragma unroll
    for (int r = 0; r < 8; ++r) {
        const float M = hmax16(m_ln[r]);
        const float w = __expf(m_ln[r] - M);
        const float L = hsum16(l_ln[r] * w);
        const float inv = 1.0f / L;
        const float s0 = hsum16(acc[r][0] * w) * inv;
        const float s1 = hsum16(acc[r][1] * w) * inv;
        const float s2 = hsum16(acc[r][2] * w) * inv;
        const float s3 = hsum16(acc[r][3] * w) * inv;
        const float outv = (lr == 0) ? s0 : (lr == 1) ? s1 : (lr == 2) ? s2 : s3;
        if (lr < 4) {
            const int row = rowTile * 16 + hg * 8 + r;
            X[((size_t)b * 64 + h * 4 + lr) * N_ + row] = outv;
        }
    }
}

// ---------------------------------------------------------------------------
// Kernel 3: y = Wo @ X + bo, BatchNorm1d(eval, running stats), LeakyReLU(0.2)
// ---------------------------------------------------------------------------
__global__ void __launch_bounds__(256)
out_kernel(const float* __restrict__ X, const float* __restrict__ Wo,
           const float* __restrict__ bo, const float* __restrict__ gamma,
           const float* __restrict__ beta, const float* __restrict__ rmean,
           const float* __restrict__ rvar, float* __restrict__ out) {
    int t = blockIdx.x;               // 0..4095
    int b = t >> 10;
    int o = (t >> 2) & 255;
    int n = ((t & 3) << 8) + threadIdx.x;

    const float* xp = X + (size_t)b * 64 * N_ + n;
    const float* wp = Wo + o * 64;    // uniform -> SMEM loads
    float acc = bo[o];
#pragma unroll 8
    for (int c = 0; c < 64; ++c)
        acc = fmaf(wp[c], xp[(size_t)c * N_], acc);

    const float g = gamma[o] * rsqrtf(rvar[o] + BN_EPS_);
    float y = fmaf(acc - rmean[o], g, beta[o]);
    y = (y > 0.0f) ? y : NEG_SLOPE_ * y;
    out[((size_t)b * HID_ + o) * N_ + n] = y;
}

extern "C" void kernel_launch(void* const* d_in, const int* in_sizes, int n_in,
                              void* d_out, int out_size, void* d_ws,
                              size_t ws_size, hipStream_t stream) {
    const float* q    = (const float*)d_in[0];
    const float* k    = (const float*)d_in[1];
    const float* v    = (const float*)d_in[2];
    const float* bias = (const float*)d_in[3];
    const float* Wq   = (const float*)d_in[4];
    const float* Wk   = (const float*)d_in[5];
    const float* Wv   = (const float*)d_in[6];
    const float* Wo   = (const float*)d_in[7];
    const float* bo   = (const float*)d_in[8];
    const float* gm   = (const float*)d_in[9];
    const float* bt   = (const float*)d_in[10];
    const float* rm   = (const float*)d_in[11];
    const float* rv   = (const float*)d_in[12];

    float* ws  = (float*)d_ws;
    float* Qh  = ws;                       // [B,H,N,4] = 262144 floats
    float* KhT = ws + 262144;              // [B,H,N,4]
    float* Vh  = ws + 524288;              // [B,H,N,4]
    float* X   = ws + 786432;              // [B,64,N]

    proj_kernel<<<3072, 256, 0, stream>>>(q, k, v, Wq, Wk, Wv, Qh, KhT, Vh);
    attn_kernel<<<512, 256, 0, stream>>>(Qh, KhT, Vh, bias, X);
    out_kernel<<<4096, 256, 0, stream>>>(X, Wo, bo, gm, bt, rm, rv,
                                         (float*)d_out);
}